// GAT_37890201485513
// MI455X (gfx1250) — compile-verified
//
#include <hip/hip_runtime.h>

typedef float v2f __attribute__((ext_vector_type(2)));
typedef float v8f __attribute__((ext_vector_type(8)));

#define NEG_SLOPE 0.2f

// ---------- order-preserving float <-> uint key (for atomic segment-max) ----------
__device__ __forceinline__ unsigned fkey(float f) {
  unsigned u = __float_as_uint(f);
  return (u & 0x80000000u) ? ~u : (u | 0x80000000u);
}
__device__ __forceinline__ float fdekey(unsigned k) {
  unsigned u = (k & 0x80000000u) ? (k & 0x7FFFFFFFu) : ~k;
  return __uint_as_float(u);
}

// ---------- zero init (grid-stride) ----------
__global__ void zero_f32(float* __restrict__ p, size_t n) {
  size_t i = (size_t)blockIdx.x * blockDim.x + threadIdx.x;
  size_t stride = (size_t)gridDim.x * blockDim.x;
  for (; i < n; i += stride) p[i] = 0.0f;
}

// ---------- C[N x Nc] = A[N x K] * B[Nc x K]^T via V_WMMA_F32_16X16X4_F32 ----------
// One wave32 per 16x16 output tile. K must be a multiple of 4.
// A-operand 16x4 layout (ISA 7.12.2): lanes 0-15 hold M=0..15 with K={0,1} in the
// two VGPRs; lanes 16-31 hold K={2,3}. B 4x16 mirrors with N in place of M.
__global__ void gemm_wmma_f32(const float* __restrict__ A, const float* __restrict__ B,
                              float* __restrict__ C, int N, int K, int Nc) {
  const int lane = threadIdx.x & 31;
  const int half = lane >> 4;     // 0: K pair {0,1}, 1: K pair {2,3}
  const int idx  = lane & 15;     // M index for A, N index for B
  const int row0 = blockIdx.x * 16;
  const int col0 = blockIdx.y * 16;

  const int ar = min(row0 + idx, N - 1);   // clamp: keep EXEC full for WMMA
  const int bc = min(col0 + idx, Nc - 1);
  const float* __restrict__ ap = A + (size_t)ar * K + 2 * half;
  const float* __restrict__ bp = B + (size_t)bc * K + 2 * half;

  v8f c = {0.f, 0.f, 0.f, 0.f, 0.f, 0.f, 0.f, 0.f};
  for (int k = 0; k < K; k += 4) {
    v2f a = *(const v2f*)(ap + k);
    v2f b = *(const v2f*)(bp + k);
    c = __builtin_amdgcn_wmma_f32_16x16x4_f32(false, a, false, b, (short)0, c,
                                              false, false);
  }

  // D layout: VGPR r, lanes 0-15 -> M=r, N=lane; lanes 16-31 -> M=8+r, N=lane-16
  const int col = col0 + idx;
  if (col < Nc) {
#pragma unroll
    for (int r = 0; r < 8; ++r) {
      const int row = row0 + r + 8 * half;
      if (row < N) C[(size_t)row * Nc + col] = c[r];
    }
  }
}

// ---------- per-(node,head) attention logits: a = <h[n,h,:], att[h,:]> ----------
__global__ void attn_logits(const float* __restrict__ h, const float* __restrict__ att_s,
                            const float* __restrict__ att_d, float* __restrict__ a_s,
                            float* __restrict__ a_d, int NH, int H, int D) {
  int i = blockIdx.x * blockDim.x + threadIdx.x;  // i = n*H + hh
  if (i >= NH) return;
  const int hh = i % H;
  const float* __restrict__ hp = h + (size_t)i * D;
  const float* __restrict__ as = att_s + (size_t)hh * D;
  const float* __restrict__ ad = att_d + (size_t)hh * D;
  float s0 = 0.f, s1 = 0.f;
  for (int dd = 0; dd < D; ++dd) {
    const float v = hp[dd];
    s0 += v * as[dd];
    s1 += v * ad[dd];
  }
  a_s[i] = s0;
  a_d[i] = s1;
}

// ---------- pass 1: segment max of leaky-relu edge logits over dst ----------
__global__ void edge_max(const int* __restrict__ src, const int* __restrict__ dst,
                         int E, int N, const float* __restrict__ a_s,
                         const float* __restrict__ a_d, unsigned* __restrict__ mkey,
                         int H) {
  const int t = blockIdx.x * blockDim.x + threadIdx.x;
  const int TE = E + N;  // edges + self loops
  if (t >= TE) return;
  const int s = (t < E) ? src[t] : (t - E);
  const int d = (t < E) ? dst[t] : (t - E);
  for (int hh = 0; hh < H; ++hh) {
    float e = a_s[s * H + hh] + a_d[d * H + hh];
    e = (e > 0.f) ? e : NEG_SLOPE * e;
    atomicMax(&mkey[d * H + hh], fkey(e));
  }
}

// ---------- pass 2: segment sum of exp(e - max) over dst ----------
__global__ void edge_expsum(const int* __restrict__ src, const int* __restrict__ dst,
                            int E, int N, const float* __restrict__ a_s,
                            const float* __restrict__ a_d,
                            const unsigned* __restrict__ mkey,
                            float* __restrict__ den, int H) {
  const int t = blockIdx.x * blockDim.x + threadIdx.x;
  const int TE = E + N;
  if (t >= TE) return;
  const int s = (t < E) ? src[t] : (t - E);
  const int d = (t < E) ? dst[t] : (t - E);
  for (int hh = 0; hh < H; ++hh) {
    float e = a_s[s * H + hh] + a_d[d * H + hh];
    e = (e > 0.f) ? e : NEG_SLOPE * e;
    const float ex = expf(e - fdekey(mkey[d * H + hh]));
    atomicAdd(&den[d * H + hh], ex);
  }
}

// ---------- pass 3: alpha-weighted scatter of h[src] into agg[dst] ----------
__global__ void edge_aggregate(const int* __restrict__ src, const int* __restrict__ dst,
                               int E, int N, const float* __restrict__ a_s,
                               const float* __restrict__ a_d,
                               const unsigned* __restrict__ mkey,
                               const float* __restrict__ den,
                               const float* __restrict__ h, float* __restrict__ agg,
                               int H, int D) {
  const int t = blockIdx.x * blockDim.x + threadIdx.x;  // over (E+N)*H
  const int TEH = (E + N) * H;
  if (t >= TEH) return;
  const int ei = t / H;
  const int hh = t - ei * H;
  const int s = (ei < E) ? src[ei] : (ei - E);
  const int d = (ei < E) ? dst[ei] : (ei - E);
  float e = a_s[s * H + hh] + a_d[d * H + hh];
  e = (e > 0.f) ? e : NEG_SLOPE * e;
  const float alpha = expf(e - fdekey(mkey[d * H + hh])) / den[d * H + hh];
  const float* __restrict__ hs = h + ((size_t)s * H + hh) * D;
  float* __restrict__ ag = agg + ((size_t)d * H + hh) * D;
  for (int dd = 0; dd < D; ++dd) atomicAdd(&ag[dd], alpha * hs[dd]);
}

// ---------- epilogues ----------
__global__ void bias_relu(const float* __restrict__ agg, const float* __restrict__ b,
                          float* __restrict__ out, int C, size_t n) {
  size_t i = (size_t)blockIdx.x * blockDim.x + threadIdx.x;
  if (i >= n) return;
  const float v = agg[i] + b[i % C];
  out[i] = (v > 0.f) ? v : 0.f;
}

__global__ void bias_add_inplace(float* __restrict__ out, const float* __restrict__ b,
                                 int C, size_t n) {
  size_t i = (size_t)blockIdx.x * blockDim.x + threadIdx.x;
  if (i >= n) return;
  out[i] += b[i % C];
}

extern "C" void kernel_launch(void* const* d_in, const int* in_sizes, int n_in,
                              void* d_out, int out_size, void* d_ws, size_t ws_size,
                              hipStream_t stream) {
  const float* x        = (const float*)d_in[0];  // [N, 256]
  const int*   ei       = (const int*)d_in[1];    // [2, E]
  const float* W1       = (const float*)d_in[2];  // [128, 256]
  const float* att_src1 = (const float*)d_in[3];  // [4, 32]
  const float* att_dst1 = (const float*)d_in[4];  // [4, 32]
  const float* b1       = (const float*)d_in[5];  // [128]
  const float* W2       = (const float*)d_in[6];  // [40, 128]
  const float* att_src2 = (const float*)d_in[7];  // [1, 40]
  const float* att_dst2 = (const float*)d_in[8];  // [1, 40]
  const float* b2       = (const float*)d_in[9];  // [40]

  const int NFEAT = 256, NHID = 128, H1 = 4, D1 = 32, NCLS = 40;
  const int N = in_sizes[0] / NFEAT;      // 50000
  const int E = in_sizes[1] / 2;          // 800000
  const int* src = ei;
  const int* dst = ei + E;

  float* out = (float*)d_out;             // [N, 40]

  // ---- workspace layout (all fp32 words) ----
  float* w = (float*)d_ws;
  float* h1   = w;                                   // [N,128]; reused as x2 later
  float* h2   = h1  + (size_t)N * NHID;              // [N,40]
  float* as1  = h2  + (size_t)N * NCLS;              // [N,4]
  float* ad1  = as1 + (size_t)N * H1;                // [N,4]
  float* as2  = ad1 + (size_t)N * H1;                // [N]
  float* ad2  = as2 + (size_t)N;                     // [N]
  float* zreg = ad2 + (size_t)N;                     // contiguous zero-init region:
  float*    agg1 = zreg;                             // [N,128]
  float*    den1 = agg1 + (size_t)N * NHID;          // [N,4]
  unsigned* m1   = (unsigned*)(den1 + (size_t)N * H1); // [N,4] keys (0 == -infty key)
  float*    den2 = (float*)(m1 + (size_t)N * H1);    // [N]
  unsigned* m2   = (unsigned*)(den2 + (size_t)N);    // [N] keys
  const size_t zcount = (size_t)N * (NHID + H1 + H1 + 1 + 1);

  const int ZB = 2048, TB = 256;
  const int TE = E + N;

  // init accumulators + output
  zero_f32<<<ZB, TB, 0, stream>>>(zreg, zcount);
  zero_f32<<<ZB, TB, 0, stream>>>(out, (size_t)N * NCLS);

  // ===== layer 1 =====
  {
    dim3 grid((N + 15) / 16, NHID / 16);
    gemm_wmma_f32<<<grid, 32, 0, stream>>>(x, W1, h1, N, NFEAT, NHID);
  }
  attn_logits<<<(N * H1 + TB - 1) / TB, TB, 0, stream>>>(h1, att_src1, att_dst1,
                                                         as1, ad1, N * H1, H1, D1);
  edge_max<<<(TE + TB - 1) / TB, TB, 0, stream>>>(src, dst, E, N, as1, ad1, m1, H1);
  edge_expsum<<<(TE + TB - 1) / TB, TB, 0, stream>>>(src, dst, E, N, as1, ad1, m1,
                                                     den1, H1);
  edge_aggregate<<<(TE * H1 + TB - 1) / TB, TB, 0, stream>>>(src, dst, E, N, as1, ad1,
                                                             m1, den1, h1, agg1, H1, D1);
  // x2 = relu(agg1 + b1); writes over h1 (no longer needed)
  float* x2 = h1;
  {
    const size_t n = (size_t)N * NHID;
    bias_relu<<<(unsigned)((n + TB - 1) / TB), TB, 0, stream>>>(agg1, b1, x2, NHID, n);
  }

  // ===== layer 2 =====
  {
    dim3 grid((N + 15) / 16, (NCLS + 15) / 16);
    gemm_wmma_f32<<<grid, 32, 0, stream>>>(x2, W2, h2, N, NHID, NCLS);
  }
  attn_logits<<<(N + TB - 1) / TB, TB, 0, stream>>>(h2, att_src2, att_dst2,
                                                    as2, ad2, N, 1, NCLS);
  edge_max<<<(TE + TB - 1) / TB, TB, 0, stream>>>(src, dst, E, N, as2, ad2, m2, 1);
  edge_expsum<<<(TE + TB - 1) / TB, TB, 0, stream>>>(src, dst, E, N, as2, ad2, m2,
                                                     den2, 1);
  edge_aggregate<<<(TE + TB - 1) / TB, TB, 0, stream>>>(src, dst, E, N, as2, ad2,
                                                        m2, den2, h2, out, 1, NCLS);
  {
    const size_t n = (size_t)N * NCLS;
    bias_add_inplace<<<(unsigned)((n + TB - 1) / TB), TB, 0, stream>>>(out, b2, NCLS, n);
  }
}